// MoEConditionalCPPN_64330020159870
// MI455X (gfx1250) — compile-verified
//
#include <hip/hip_runtime.h>
#include <hip/hip_bf16.h>

typedef float v2f __attribute__((ext_vector_type(2)));
typedef float v8f __attribute__((ext_vector_type(8)));

#define NPTS 262144
#define D_INF 7
#define D_HID 12
#define NEXP 6
#define NLAY 6
#define R_HID 48

__device__ __forceinline__ void wavebar() {
#if __has_builtin(__builtin_amdgcn_wave_barrier)
  __builtin_amdgcn_wave_barrier();
#else
  __syncthreads();
#endif
}

__device__ __forceinline__ float fast_rcp(float v) {
#if __has_builtin(__builtin_amdgcn_rcpf)
  return __builtin_amdgcn_rcpf(v);
#else
  return 1.0f / v;
#endif
}

// branch-free tanh: 1 - 2/(e^{2x}+1). Saturates correctly at +/-inf.
__device__ __forceinline__ float tanh_fast(float v) {
  float e = __expf(2.0f * v);
  return 1.0f - 2.0f * fast_rcp(e + 1.0f);
}

// D(16x16,f32) += A(16x4,f32) x B(4x16,f32)
__device__ __forceinline__ v8f wmma4(v2f a, v2f b, v8f c) {
  return __builtin_amdgcn_wmma_f32_16x16x4_f32(false, a, false, b, (short)0, c,
                                               false, false);
}

// activation (branchless): features 0-7,10 identity, 8-9 gaussian, 11 sine.
// 'n' is this lane's feature column in the D layout; >=12 stay zero.
__device__ __forceinline__ v8f actv(v8f d, int n) {
  const bool isg = (n == 8) || (n == 9);
  const bool iss = (n == 11);
  v8f h;
#pragma unroll
  for (int r = 0; r < 8; r++) {
    float g = __expf(-d[r] * d[r]) * 2.0f - 1.0f;
    float s = __sinf(d[r]);
    float v = isg ? g : d[r];
    h[r] = iss ? s : v;
  }
  return h;
}

__global__ __launch_bounds__(256, 2) void cppn_moe_kernel(
    const float* __restrict__ x, const float* __restrict__ Wr1,
    const float* __restrict__ Wr2, const float* __restrict__ We_in,
    const float* __restrict__ We_hid, const float* __restrict__ We_out,
    float* __restrict__ out) {
  // staging: routing tanh features (cols 0..47); cols 0..15 reused for the
  // hidden-state D->A transpose in the expert phase (phases don't overlap).
  __shared__ float lds_s[8][16][R_HID];
  __shared__ float lds_w[8][16][NEXP];  // per-point softmax weights
  // weights, preloaded once per block, zero-padded to 16 columns:
  __shared__ float sWr1[4 * R_HID];        // [4][48]
  __shared__ float sWr2[R_HID * 16];       // [48][16], cols 6..15 = 0
  __shared__ float sWin[NEXP * 8 * 16];    // [e][8][16], row7/cols>=12 = 0
  __shared__ float sWhid[5 * NEXP * D_HID * 16];  // [l*6+e][12][16]
  __shared__ float sWout[NEXP * D_HID * 16];      // [e][12][16], cols>=3 = 0

  const int tid = threadIdx.x;

  // ---- cooperative, padded weight preload ----
  for (int i = tid; i < 4 * R_HID; i += 256) sWr1[i] = Wr1[i];
  for (int i = tid; i < R_HID * 16; i += 256) {
    int k = i >> 4, c = i & 15;
    sWr2[i] = (c < NEXP) ? Wr2[k * NEXP + c] : 0.f;
  }
  for (int i = tid; i < NEXP * 8 * 16; i += 256) {
    int e = i >> 7, rc = i & 127, r = rc >> 4, c = rc & 15;
    sWin[i] = (r < D_INF && c < D_HID) ? We_in[(e * D_INF + r) * D_HID + c] : 0.f;
  }
  for (int i = tid; i < 5 * NEXP * D_HID * 16; i += 256) {
    int le = i / 192, rc = i % 192, r = rc >> 4, c = rc & 15;
    sWhid[i] = (c < D_HID) ? We_hid[(le * D_HID + r) * D_HID + c] : 0.f;
  }
  for (int i = tid; i < NEXP * D_HID * 16; i += 256) {
    int e = i / 192, rc = i % 192, r = rc >> 4, c = rc & 15;
    sWout[i] = (c < 3) ? We_out[(e * D_HID + r) * 3 + c] : 0.f;
  }
  __syncthreads();

  const int wave = tid >> 5;
  const int lane = tid & 31;
  const int half = lane >> 4;   // 0: K even pair / D-rows 0-7, 1: odd / rows 8-15
  const int nidx = lane & 15;   // column (B/C/D) or row (A) index
  const int m0 = half << 3;     // first D-row held by this lane
  const int tile = (blockIdx.x * 8 + wave) * 16;
  const int prow = tile + nidx; // the point whose A-row this lane owns
  const int kk = 2 * half;      // base K inside a K=4 chunk for this lane

  // ---- A operands for the input layer: x tile 16x7 padded to 16x8 ----
  v2f Ax0, Ax1;
  {
    Ax0.x = x[prow * D_INF + kk];
    Ax0.y = x[prow * D_INF + kk + 1];
    const int k1 = 4 + kk;
    Ax1.x = (k1 < D_INF) ? x[prow * D_INF + k1] : 0.f;
    Ax1.y = (k1 + 1 < D_INF) ? x[prow * D_INF + k1 + 1] : 0.f;
  }

  // =================== routing, on the matrix path ===================
  {
    // T(16x48) = tanh(spatial(16x4) @ Wr1(4x48)) : 3 column-tile WMMAs
    v8f T[3];
#pragma unroll
    for (int t = 0; t < 3; t++) {
      v2f B;
      B.x = sWr1[kk * R_HID + 16 * t + nidx];
      B.y = sWr1[(kk + 1) * R_HID + 16 * t + nidx];
      v8f z = {};
      T[t] = wmma4(Ax0, B, z);
    }
#pragma unroll
    for (int t = 0; t < 3; t++)
#pragma unroll
      for (int r = 0; r < 8; r++) T[t][r] = tanh_fast(T[t][r]);
#pragma unroll
    for (int t = 0; t < 3; t++)
#pragma unroll
      for (int r = 0; r < 8; r++) lds_s[wave][m0 + r][16 * t + nidx] = T[t][r];
    wavebar();

    // logits(16x6) = T(16x48) @ Wr2(48x16pad) : 12 accumulating K-chunks
    v8f L = {};
#pragma unroll
    for (int c = 0; c < 12; c++) {
      const int k = 4 * c + kk;
      v2f A, B;
      A.x = lds_s[wave][nidx][k];
      A.y = lds_s[wave][nidx][k + 1];
      B.x = sWr2[k * 16 + nidx];
      B.y = sWr2[(k + 1) * 16 + nidx];
      L = wmma4(A, B, L);
    }
    if (nidx < NEXP) {
#pragma unroll
      for (int r = 0; r < 8; r++) lds_w[wave][m0 + r][nidx] = L[r];
    }
    wavebar();

    // per-point softmax on lanes 0-15; emit combined[0] = x * sum(w)
    if (half == 0) {
      float lg[NEXP];
#pragma unroll
      for (int e = 0; e < NEXP; e++) lg[e] = lds_w[wave][nidx][e];
      float mx = lg[0];
#pragma unroll
      for (int e = 1; e < NEXP; e++) mx = fmaxf(mx, lg[e]);
      float s = 0.f, w[NEXP];
#pragma unroll
      for (int e = 0; e < NEXP; e++) { w[e] = __expf(lg[e] - mx); s += w[e]; }
      const float inv = fast_rcp(s);
      float ws = 0.f;
#pragma unroll
      for (int e = 0; e < NEXP; e++) { w[e] *= inv; ws += w[e]; }
#pragma unroll
      for (int e = 0; e < NEXP; e++) lds_w[wave][nidx][e] = w[e];
      const size_t b0 = (size_t)3 * NPTS + (size_t)prow * D_INF;
#pragma unroll
      for (int c = 0; c < D_INF; c++) out[b0 + c] = x[prow * D_INF + c] * ws;
    }
    wavebar();
  }

  // =================== expert MLPs ===================
  v8f acc[NLAY];
#pragma unroll
  for (int l = 0; l < NLAY; l++) { v8f z = {}; acc[l] = z; }
  v8f accO = {};

  for (int e = 0; e < NEXP; e++) {
    // per-row routing weight vector (rows m0..m0+7 of this tile)
    v8f wv;
#pragma unroll
    for (int r = 0; r < 8; r++) wv[r] = lds_w[wave][m0 + r][e];

    // ---- input layer: (16x8) x (8x16), padded We_in[e] ----
    v8f d = {};
    {
      const float* W = sWin + e * 128;  // [8][16]
      v2f B0, B1;
      B0.x = W[kk * 16 + nidx];
      B0.y = W[(kk + 1) * 16 + nidx];
      B1.x = W[(4 + kk) * 16 + nidx];
      B1.y = W[(5 + kk) * 16 + nidx];
      d = wmma4(Ax0, B0, d);
      d = wmma4(Ax1, B1, d);
    }
    v8f h = actv(d, nidx);
    acc[0] += wv * h;

    // ---- 5 hidden layers: (16x12) x (12x16pad) = 3 chained WMMAs each ----
#pragma unroll
    for (int l = 0; l < NLAY - 1; l++) {
#pragma unroll
      for (int r = 0; r < 8; r++) lds_s[wave][m0 + r][nidx] = h[r];
      wavebar();
      const float* W = sWhid + (l * NEXP + e) * 192;  // [12][16]
      v8f d2 = {};
#pragma unroll
      for (int c = 0; c < 3; c++) {
        const int k = 4 * c + kk;
        v2f A, B;
        A.x = lds_s[wave][nidx][k];
        A.y = lds_s[wave][nidx][k + 1];
        B.x = W[k * 16 + nidx];
        B.y = W[(k + 1) * 16 + nidx];
        d2 = wmma4(A, B, d2);
      }
      h = actv(d2, nidx);
      acc[l + 1] += wv * h;
    }

    // ---- output layer: (16x12) x (12x16pad), no activation ----
    {
#pragma unroll
      for (int r = 0; r < 8; r++) lds_s[wave][m0 + r][nidx] = h[r];
      wavebar();
      const float* W = sWout + e * 192;  // [12][16]
      v8f d3 = {};
#pragma unroll
      for (int c = 0; c < 3; c++) {
        const int k = 4 * c + kk;
        v2f A, B;
        A.x = lds_s[wave][nidx][k];
        A.y = lds_s[wave][nidx][k + 1];
        B.x = W[k * 16 + nidx];
        B.y = W[(k + 1) * 16 + nidx];
        d3 = wmma4(A, B, d3);
      }
      accO += wv * d3;
    }
  }

  // =================== outputs ===================
  // combined[1..6]: (N,12) each, at offsets (10 + 12*l) * N
  if (nidx < D_HID) {
#pragma unroll
    for (int l = 0; l < NLAY; l++) {
      const size_t base = (size_t)(10 + 12 * l) * NPTS;
#pragma unroll
      for (int r = 0; r < 8; r++)
        out[base + (size_t)(tile + m0 + r) * D_HID + nidx] = acc[l][r];
    }
  }
  // combined[7]: (N,3) at 82*N; hsv channels at c*N
  if (nidx < 3) {
    const size_t base7 = (size_t)82 * NPTS;
#pragma unroll
    for (int r = 0; r < 8; r++) {
      const int p = tile + m0 + r;
      const float v = accO[r];
      out[base7 + (size_t)p * 3 + nidx] = v;
      out[(size_t)nidx * NPTS + p] = v;
    }
  }
}

extern "C" void kernel_launch(void* const* d_in, const int* in_sizes, int n_in,
                              void* d_out, int out_size, void* d_ws,
                              size_t ws_size, hipStream_t stream) {
  const float* x      = (const float*)d_in[0];
  const float* Wr1    = (const float*)d_in[1];
  const float* Wr2    = (const float*)d_in[2];
  const float* We_in  = (const float*)d_in[3];
  const float* We_hid = (const float*)d_in[4];
  const float* We_out = (const float*)d_in[5];
  float* out = (float*)d_out;

  dim3 grid(NPTS / 128);  // 8 waves/block * 16 points/wave
  dim3 block(256);
  hipLaunchKernelGGL(cppn_moe_kernel, grid, block, 0, stream, x, Wr1, Wr2,
                     We_in, We_hid, We_out, out);
}